// MoAT_31044023616148
// MI455X (gfx1250) — compile-verified
//
#include <hip/hip_runtime.h>
#include <math.h>

// ---------------------------------------------------------------------------
// MoAT reference pipeline for MI455X (gfx1250, wave32).
//   n=1024, l=4, batch=32, 40 Sinkhorn iterations.
// Heavy phase: 33 unpivoted blocked LU factorizations (log-det, NB=32),
// trailing updates on V_WMMA_F32_16X16X4_F32 (fp32 matrix cores; fp32 is
// required for log-det accuracy). Working set (33 x 4MB) fits in 192MB L2.
// Unpivoted LU is stable here: L[1:,1:] is a principal submatrix of a
// nonneg-weight Laplacian => strictly diagonally dominant.
// Workspace requirement: ~139 MB.
// ---------------------------------------------------------------------------

#define N      1024
#define MATS   (1024ull * 1024ull)   // floats per LU matrix
#define NMAT   33                    // 32 batch Laplacians + L0
#define NB     32                    // LU block size (K depth per update step)
#define NSTEP  (N / NB)              // 32 block steps
#define EPS_S  1e-7f

typedef float v2f __attribute__((ext_vector_type(2)));
typedef float v8f __attribute__((ext_vector_type(8)));

__device__ __forceinline__ float frcp(float x) { return __builtin_amdgcn_rcpf(x); }

// ---------------------------------------------------------------- V softmax
__global__ void k_softmax_v(const float* __restrict__ Vc, float* __restrict__ Vm) {
    int i = blockIdx.x * blockDim.x + threadIdx.x;
    if (i >= N) return;
    float v0 = Vc[i*4+0], v1 = Vc[i*4+1], v2 = Vc[i*4+2], v3 = Vc[i*4+3];
    float mx = fmaxf(fmaxf(v0, v1), fmaxf(v2, v3));
    float e0 = expf(v0-mx), e1 = expf(v1-mx), e2 = expf(v2-mx), e3 = expf(v3-mx);
    float inv = 1.0f / (e0+e1+e2+e3);
    Vm[i*4+0] = e0*inv; Vm[i*4+1] = e1*inv; Vm[i*4+2] = e2*inv; Vm[i*4+3] = e3*inv;
}

// ------------------------------------------- Ws (sym tril sigmoid) + L0 body
__global__ void k_ws(const float* __restrict__ W, float* __restrict__ WsM,
                     float* __restrict__ Abase, float* __restrict__ col0) {
    int i = blockIdx.x;
    float* A32 = Abase + 32ull * MATS;
    for (int j = threadIdx.x; j < N; j += blockDim.x) {
        float w;
        if (i > j)      w = 1.0f / (1.0f + expf(-W[(size_t)i*N + j]));
        else if (i < j) w = 1.0f / (1.0f + expf(-W[(size_t)j*N + i]));
        else            w = 0.0f;
        WsM[(size_t)i*N + j] = w;
        if (i >= 1 && j >= 1) A32[(size_t)(i-1)*N + (j-1)] = -w;
        if (j == 0) col0[32*N + i] = w;
    }
}

// ----------------------------------------------------- Pr gather, invPr, y
__global__ void k_pr(const int* __restrict__ x, const float* __restrict__ Vm,
                     float* __restrict__ invPr, float* __restrict__ yv) {
    __shared__ float red[256];
    int b = blockIdx.x;
    float acc = 0.0f;
    for (int i = threadIdx.x; i < N; i += 256) {
        int   xi = x[b*N + i];
        float pr = Vm[i*4 + xi];
        invPr[b*N + i] = 1.0f / pr;       // IEEE div: feeds P directly
        acc += logf(pr + 1e-7f);
    }
    red[threadIdx.x] = acc;
    __syncthreads();
    for (int s = 128; s > 0; s >>= 1) {
        if (threadIdx.x < s) red[threadIdx.x] += red[threadIdx.x + s];
        __syncthreads();
    }
    if (threadIdx.x == 0) yv[b] = red[0];
}

// --------------------- per-pair 4x4 Sinkhorn in registers + Laplacian scatter
// Block handles 256 consecutive j for one uniform i (scalar x[b,i], V[i,:]).
__global__ void k_sinkhorn(const float* __restrict__ Ec, const int* __restrict__ x,
                           const float* __restrict__ Vm, const float* __restrict__ invPr,
                           const float* __restrict__ WsM, float* __restrict__ Abase,
                           float* __restrict__ col0) {
    int i = blockIdx.x >> 2;                               // uniform per block
    int j = ((blockIdx.x & 3) << 8) | threadIdx.x;
    size_t gid = (size_t)i * N + j;

    float e[16];
    const float4* src = reinterpret_cast<const float4*>(Ec + gid * 16);
    float s = 0.0f;
#pragma unroll
    for (int q = 0; q < 4; ++q) {
        float4 v = src[q];
        e[q*4+0] = expf(v.x); e[q*4+1] = expf(v.y);
        e[q*4+2] = expf(v.z); e[q*4+3] = expf(v.w);
        s += e[q*4+0] + e[q*4+1] + e[q*4+2] + e[q*4+3];
    }
    float inv0 = frcp(s);
#pragma unroll
    for (int t = 0; t < 16; ++t) e[t] *= inv0;

    float a4[4], b4[4];
#pragma unroll
    for (int c = 0; c < 4; ++c) { a4[c] = Vm[i*4 + c]; b4[c] = Vm[j*4 + c]; }

    for (int it = 0; it < 40; ++it) {
#pragma unroll
        for (int mr = 0; mr < 4; ++mr) {                   // row marginals -> V[i]
            float r = e[mr*4] + e[mr*4+1] + e[mr*4+2] + e[mr*4+3];
            float f = a4[mr] * frcp(r + EPS_S);
            e[mr*4] *= f; e[mr*4+1] *= f; e[mr*4+2] *= f; e[mr*4+3] *= f;
        }
#pragma unroll
        for (int nc = 0; nc < 4; ++nc) {                   // col marginals -> V[j]
            float cs = e[nc] + e[4+nc] + e[8+nc] + e[12+nc];
            float g = b4[nc] * frcp(cs + EPS_S);
            e[nc] *= g; e[4+nc] *= g; e[8+nc] *= g; e[12+nc] *= g;
        }
        float tot = 0.0f;
#pragma unroll
        for (int t = 0; t < 16; ++t) tot += e[t];
        float h = frcp(tot + EPS_S);
#pragma unroll
        for (int t = 0; t < 16; ++t) e[t] *= h;
    }

    float wij  = WsM[gid];
    bool  diag = (i == j);
#pragma unroll 4
    for (int b = 0; b < 32; ++b) {
        int xi = x[b*N + i];                               // scalar (uniform i)
        int xj = x[b*N + j];                               // coalesced
        float r0, r1, r2, r3;
        switch (xi) {                                      // uniform row select
            case 0:  r0 = e[0];  r1 = e[1];  r2 = e[2];  r3 = e[3];  break;
            case 1:  r0 = e[4];  r1 = e[5];  r2 = e[6];  r3 = e[7];  break;
            case 2:  r0 = e[8];  r1 = e[9];  r2 = e[10]; r3 = e[11]; break;
            default: r0 = e[12]; r1 = e[13]; r2 = e[14]; r3 = e[15]; break;
        }
        float ev = (xj == 0) ? r0 : (xj == 1) ? r1 : (xj == 2) ? r2 : r3;
        if (diag) ev = 0.0f;
        ev = fminf(fmaxf(ev, 0.0f), 1.0f);
        float wp = wij * ev * invPr[b*N + i] * invPr[b*N + j];
        if (i >= 1 && j >= 1)
            Abase[(size_t)b*MATS + (size_t)(i-1)*N + (j-1)] = -wp;   // off-diag
        if (j == 0) col0[b*N + i] = wp;                              // dropped col
    }
}

// ------- finalize rows: diag = rowsum(WP)+eps, +eps elsewhere, identity pad
__global__ void k_finalize(float* __restrict__ Abase, const float* __restrict__ col0) {
    __shared__ float red[256];
    int m = blockIdx.x >> 10;
    int r = blockIdx.x & 1023;
    float* row = Abase + (size_t)m*MATS + (size_t)r*N;
    if (r == 1023) {                 // padding row/col -> identity (log 1 = 0)
        for (int c = threadIdx.x; c < N; c += 256) row[c] = (c == 1023) ? 1.0f : 0.0f;
        return;
    }
    float s = 0.0f;
    for (int c = threadIdx.x; c < 1023; c += 256) s += row[c];   // = -sum_offdiag WP
    red[threadIdx.x] = s;
    __syncthreads();
    for (int st = 128; st > 0; st >>= 1) {
        if (threadIdx.x < st) red[threadIdx.x] += red[threadIdx.x + st];
        __syncthreads();
    }
    float rowsum_wp = col0[m*N + (r+1)] - red[0];
    for (int c = threadIdx.x; c < N; c += 256) {
        float v;
        if (c == 1023)     v = 0.0f;
        else if (c == r)   v = rowsum_wp + 1e-7f;
        else               v = row[c] + 1e-7f;
        row[c] = v;
    }
}

// -------------------- LU panel: 32x32 diag factor (LDS) + triangular solves
__global__ void k_panel(float* __restrict__ Abase, float* __restrict__ ldpart, int kb) {
    __shared__ float D[NB * NB];                 // 4 KB
    int m  = blockIdx.x;
    float* A = Abase + (size_t)m * MATS;
    int k0  = kb * NB;
    int tid = threadIdx.x;                       // 512 threads

#pragma unroll
    for (int q = 0; q < 2; ++q) {                // load 32x32 block
        int cell = tid + 512*q;
        int r = cell >> 5, c = cell & 31;
        D[cell] = A[(size_t)(k0 + r)*N + k0 + c];
    }
    __syncthreads();

    for (int p = 0; p < NB; ++p) {               // unpivoted (diag dominant)
        if (tid > p && tid < NB) D[tid*NB + p] *= frcp(D[p*NB + p]);
        __syncthreads();
#pragma unroll
        for (int q = 0; q < 2; ++q) {
            int cell = tid + 512*q;
            int r = cell >> 5, c = cell & 31;
            if (r > p && c > p) D[cell] -= D[r*NB + p] * D[p*NB + c];
        }
        __syncthreads();
    }

    if (tid == 0) {
        float s = 0.0f;
#pragma unroll
        for (int p = 0; p < NB; ++p) s += logf(fabsf(D[p*NB + p]));
        ldpart[m*NSTEP + kb] = s;                // partial log|det|
    }

    int base = k0 + NB;
    // U12 = L11^{-1} A12 : one column per thread
    for (int j = base + tid; j < N; j += blockDim.x) {
        float xc[NB];
#pragma unroll
        for (int r = 0; r < NB; ++r) {
            float v = A[(size_t)(k0 + r)*N + j];
#pragma unroll
            for (int t = 0; t < r; ++t) v -= D[r*NB + t] * xc[t];
            xc[r] = v;
            A[(size_t)(k0 + r)*N + j] = v;
        }
    }
    // L21 = A21 U11^{-1} : one row per thread
    for (int i2 = base + tid; i2 < N; i2 += blockDim.x) {
        float xr[NB];
#pragma unroll
        for (int c = 0; c < NB; ++c) {
            float v = A[(size_t)i2*N + k0 + c];
#pragma unroll
            for (int t = 0; t < c; ++t) v -= xr[t] * D[t*NB + c];
            v *= frcp(D[c*NB + c]);
            xr[c] = v;
            A[(size_t)i2*N + k0 + c] = v;
        }
    }
}

// ----- trailing update: A22 -= L21*U12 (K=32), one 16x16 tile/wave via WMMA
__global__ void k_update(float* __restrict__ Abase, int kb) {
    int T = (N - NB * (kb + 1)) >> 4;            // 16-wide tiles per side
    int m = blockIdx.x;
    float* A = Abase + (size_t)m * MATS;
    int wave = threadIdx.x >> 5;                 // wave32
    int lane = threadIdx.x & 31;
    int t = blockIdx.y * 8 + wave;
    if (t >= T * T) return;                      // wave-uniform: EXEC stays full
    int bi = t / T, bj = t - bi * T;
    int k0 = kb * NB;
    int i0 = k0 + NB + bi * 16;
    int j0 = k0 + NB + bj * 16;

    int nn   = lane & 15;                        // N index (B/C/D layout)
    int half = lane >> 4;
    int mh   = half * 8;                         // C/D row offset per lane-half
    int kh   = half * 2;                         // A/B K offset per lane-half
    int mrow = i0 + (lane & 15);                 // A-matrix M index

    v8f c;
#pragma unroll
    for (int v = 0; v < 8; ++v) c[v] = A[(size_t)(i0 + mh + v)*N + j0 + nn];

#pragma unroll
    for (int kk = 0; kk < NB/4; ++kk) {          // K=32 as 8 chained K=4 WMMAs
        int kc = k0 + kk*4 + kh;
        v2f a, b;
        a[0] = -A[(size_t)mrow*N + kc];          // negate L21 -> C - L21*U12
        a[1] = -A[(size_t)mrow*N + kc + 1];
        b[0] =  A[(size_t)kc      *N + j0 + nn];
        b[1] =  A[(size_t)(kc + 1)*N + j0 + nn];
        c = __builtin_amdgcn_wmma_f32_16x16x4_f32(false, a, false, b,
                                                  (short)0, c, false, false);
    }
#pragma unroll
    for (int v = 0; v < 8; ++v) A[(size_t)(i0 + mh + v)*N + j0 + nn] = c[v];
}

// ------------------------------------------------------------- combine out
__global__ void k_out(const float* __restrict__ yv, const float* __restrict__ ldpart,
                      float* __restrict__ out) {
    int b = threadIdx.x;
    if (b >= 32) return;
    float ld1 = 0.0f, ld0 = 0.0f;
    for (int k = 0; k < NSTEP; ++k) {
        ld1 += ldpart[b*NSTEP + k];
        ld0 += ldpart[32*NSTEP + k];
    }
    out[b] = yv[b] + ld1 - ld0;
}

// ---------------------------------------------------------------------------
extern "C" void kernel_launch(void* const* d_in, const int* in_sizes, int n_in,
                              void* d_out, int out_size, void* d_ws, size_t ws_size,
                              hipStream_t stream) {
    const int*   x  = (const int*)  d_in[0];   // (32,1024)
    const float* W  = (const float*)d_in[1];   // (1024,1024)
    const float* Vc = (const float*)d_in[2];   // (1024,4)
    const float* Ec = (const float*)d_in[3];   // (1,1024,1024,4,4)
    float* out = (float*)d_out;                // (32,)

    float* ws     = (float*)d_ws;              // needs ~139 MB
    float* A      = ws;                        // 33 * 1024*1024
    float* WsM    = A     + NMAT * MATS;       // 1024*1024
    float* Vm     = WsM   + MATS;              // 1024*4
    float* invPr  = Vm    + 4096;              // 32*1024
    float* yv     = invPr + 32768;             // 32
    float* col0   = yv    + 32;                // 33*1024
    float* ldpart = col0  + NMAT * 1024;       // 33*NSTEP

    k_softmax_v<<<4, 256, 0, stream>>>(Vc, Vm);
    k_ws       <<<1024, 256, 0, stream>>>(W, WsM, A, col0);
    k_pr       <<<32, 256, 0, stream>>>(x, Vm, invPr, yv);
    k_sinkhorn <<<4096, 256, 0, stream>>>(Ec, x, Vm, invPr, WsM, A, col0);
    k_finalize <<<NMAT * 1024, 256, 0, stream>>>(A, col0);

    for (int kb = 0; kb < NSTEP; ++kb) {
        k_panel<<<NMAT, 512, 0, stream>>>(A, ldpart, kb);
        int T = (N - NB * (kb + 1)) >> 4;
        if (T > 0) {
            dim3 grid(NMAT, (unsigned)((T * T + 7) / 8));
            k_update<<<grid, 256, 0, stream>>>(A, kb);
        }
    }
    k_out<<<1, 32, 0, stream>>>(yv, ldpart, out);
}